// AutoencoderGAT_GCN_30081950941403
// MI455X (gfx1250) — compile-verified
//
#include <hip/hip_runtime.h>
#include <math.h>

typedef float v2f __attribute__((ext_vector_type(2)));
typedef float v8f __attribute__((ext_vector_type(8)));

#define HEADS 2
#define CHAN  512
#define HC    1024   // HEADS*CHAN

// ---------------------------------------------------------------- utilities
__global__ void fill_f32(float* __restrict__ p, float v, int count) {
  int i = blockIdx.x * blockDim.x + threadIdx.x;
  if (i < count) p[i] = v;
}

__global__ void bias_act(float* __restrict__ X, const float* __restrict__ b,
                         int total, int cols, int relu) {
  int i = blockIdx.x * blockDim.x + threadIdx.x;
  if (i >= total) return;
  float v = X[i] + b[i % cols];
  X[i] = relu ? fmaxf(v, 0.0f) : v;
}

// ------------------------------------------------- fp32 WMMA GEMM (16x16x4)
// C[M,N] = A[M,K] @ B[K,N]; requires M%16==0, N%64==0, K%4==0.
// One wave (32 threads) computes a 16x64 tile: 4 accumulators reuse A frag.
// A frag (16x4 f32): lanes 0-15 rows M0..M0+15, lane-half selects K {0,1}/{2,3}.
// B frag (4x16 f32): lanes 0-15 cols, VGPR0/1 = K rows {kh, kh+1}.
__global__ void gemm_f32_wmma(const float* __restrict__ A, const float* __restrict__ B,
                              float* __restrict__ C, int M, int N, int K) {
  const int lane = threadIdx.x;
  const int m0 = blockIdx.x * 16;
  const int n0 = blockIdx.y * 64;
  const int row = m0 + (lane & 15);
  const int kh  = (lane >> 4) << 1;           // 0 or 2
  const int col = n0 + (lane & 15);
  v8f acc0 = {}, acc1 = {}, acc2 = {}, acc3 = {};
  const float* ap = A + (size_t)row * K + kh;
  for (int k = 0; k < K; k += 4) {
    v2f a; a[0] = ap[k]; a[1] = ap[k + 1];
    const float* bp = B + (size_t)(k + kh) * N + col;
    v2f b0; b0[0] = bp[0];  b0[1] = bp[N];
    v2f b1; b1[0] = bp[16]; b1[1] = bp[N + 16];
    v2f b2; b2[0] = bp[32]; b2[1] = bp[N + 32];
    v2f b3; b3[0] = bp[48]; b3[1] = bp[N + 48];
    acc0 = __builtin_amdgcn_wmma_f32_16x16x4_f32(false, a, false, b0, (short)0, acc0, false, false);
    acc1 = __builtin_amdgcn_wmma_f32_16x16x4_f32(false, a, false, b1, (short)0, acc1, false, false);
    acc2 = __builtin_amdgcn_wmma_f32_16x16x4_f32(false, a, false, b2, (short)0, acc2, false, false);
    acc3 = __builtin_amdgcn_wmma_f32_16x16x4_f32(false, a, false, b3, (short)0, acc3, false, false);
  }
  const int mr = m0 + ((lane >> 4) << 3);     // row base for this lane half
  const int nn = lane & 15;
#pragma unroll
  for (int r = 0; r < 8; ++r) {
    float* cp = C + (size_t)(mr + r) * N + n0 + nn;
    cp[0]  = acc0[r];
    cp[16] = acc1[r];
    cp[32] = acc2[r];
    cp[48] = acc3[r];
  }
}

// ---------------------------------------------------------------- GAT parts
__device__ __forceinline__ unsigned fkey(float f) {
  unsigned b = __float_as_uint(f);
  return (b & 0x80000000u) ? ~b : (b | 0x80000000u);
}
__device__ __forceinline__ float funkey(unsigned k) {
  unsigned b = (k & 0x80000000u) ? (k & 0x7FFFFFFFu) : ~k;
  return __uint_as_float(b);
}
__device__ __forceinline__ float lrelu02(float v) { return v > 0.0f ? v : 0.2f * v; }

// alS[n,h] = dot(H[n, h*CHAN:...], aS[h]);  one wave per (node,head)
__global__ void row_dot2(const float* __restrict__ Hm, const float* __restrict__ aS,
                         const float* __restrict__ aD, float* __restrict__ alS,
                         float* __restrict__ alD, int n) {
  int wave = (blockIdx.x * blockDim.x + threadIdx.x) >> 5;
  int lane = threadIdx.x & 31;
  if (wave >= n * HEADS) return;
  int node = wave >> 1, head = wave & 1;
  const float* hp = Hm + (size_t)node * HC + head * CHAN;
  const float* as = aS + head * CHAN;
  const float* ad = aD + head * CHAN;
  float ss = 0.0f, sd = 0.0f;
  for (int c = lane; c < CHAN; c += 32) {
    float hv = hp[c];
    ss += hv * as[c];
    sd += hv * ad[c];
  }
  for (int o = 16; o; o >>= 1) {
    ss += __shfl_xor(ss, o, 32);
    sd += __shfl_xor(sd, o, 32);
  }
  if (lane == 0) { alS[wave] = ss; alD[wave] = sd; }
}

// pass 1: segment max of leaky_relu scores via ordered-uint atomicMax
__global__ void gat_emax(const int* __restrict__ src, const int* __restrict__ dst,
                         const float* __restrict__ alS, const float* __restrict__ alD,
                         unsigned* __restrict__ emaxB, int E, int n) {
  int tid = blockIdx.x * blockDim.x + threadIdx.x;
  int total = (E + n) * HEADS;
  if (tid >= total) return;
  int eid = tid >> 1, h = tid & 1;
  int s, d;
  if (eid < E) { s = src[eid]; d = dst[eid]; } else { s = d = eid - E; }
  float e = lrelu02(alS[s * HEADS + h] + alD[d * HEADS + h]);
  atomicMax(&emaxB[d * HEADS + h], fkey(e));
}

// pass 2: segment sum of exp(e - emax)
__global__ void gat_esum(const int* __restrict__ src, const int* __restrict__ dst,
                         const float* __restrict__ alS, const float* __restrict__ alD,
                         const unsigned* __restrict__ emaxB, float* __restrict__ esum,
                         int E, int n) {
  int tid = blockIdx.x * blockDim.x + threadIdx.x;
  int total = (E + n) * HEADS;
  if (tid >= total) return;
  int eid = tid >> 1, h = tid & 1;
  int s, d;
  if (eid < E) { s = src[eid]; d = dst[eid]; } else { s = d = eid - E; }
  float e = lrelu02(alS[s * HEADS + h] + alD[d * HEADS + h]);
  float em = funkey(emaxB[d * HEADS + h]);
  atomicAdd(&esum[d * HEADS + h], expf(e - em));
}

// pass 3: out[d] += h[s] * alpha ; one block per (real or self-loop) edge
__global__ void gat_agg(const int* __restrict__ src, const int* __restrict__ dst,
                        const float* __restrict__ Hm, const float* __restrict__ alS,
                        const float* __restrict__ alD, const unsigned* __restrict__ emaxB,
                        const float* __restrict__ esum, float* __restrict__ out, int E) {
  int eid = blockIdx.x;
  int s, d;
  if (eid < E) { s = src[eid]; d = dst[eid]; } else { s = d = eid - E; }
  float alpha[HEADS];
#pragma unroll
  for (int h = 0; h < HEADS; ++h) {
    float e = lrelu02(alS[s * HEADS + h] + alD[d * HEADS + h]);
    float em = funkey(emaxB[d * HEADS + h]);
    float ex = expf(e - em);
    alpha[h] = ex / (esum[d * HEADS + h] + 1e-16f);
  }
  const float* hp = Hm + (size_t)s * HC;
  float* op = out + (size_t)d * HC;
  for (int c = threadIdx.x; c < HC; c += blockDim.x)
    atomicAdd(&op[c], hp[c] * alpha[c >> 9]);   // c>>9 == c/CHAN
}

// ---------------------------------------------------------------- GCN parts
__global__ void deg_accum(const int* __restrict__ dst, float* __restrict__ deg, int E) {
  int i = blockIdx.x * blockDim.x + threadIdx.x;
  if (i < E) atomicAdd(&deg[dst[i]], 1.0f);
}
__global__ void deg_to_dinv(float* __restrict__ deg, int n) {
  int i = blockIdx.x * blockDim.x + threadIdx.x;
  if (i < n) { float d = deg[i]; deg[i] = d > 0.0f ? rsqrtf(d) : 0.0f; }
}
__global__ void gcn_agg(const int* __restrict__ src, const int* __restrict__ dst,
                        const float* __restrict__ Hm, const float* __restrict__ dinv,
                        float* __restrict__ out, int E, int cols) {
  int eid = blockIdx.x;
  int s, d;
  if (eid < E) { s = src[eid]; d = dst[eid]; } else { s = d = eid - E; }
  float norm = dinv[s] * dinv[d];
  const float* hp = Hm + (size_t)s * cols;
  float* op = out + (size_t)d * cols;
  for (int c = threadIdx.x; c < cols; c += blockDim.x)
    atomicAdd(&op[c], hp[c] * norm);
}

// ---------------------------------------------------------------- pdist
__global__ void rowsq(const float* __restrict__ Dm, float* __restrict__ sq, int n, int K) {
  int wave = (blockIdx.x * blockDim.x + threadIdx.x) >> 5;
  int lane = threadIdx.x & 31;
  if (wave >= n) return;
  const float* p = Dm + (size_t)wave * K;
  float s = 0.0f;
  for (int c = lane; c < K; c += 32) { float v = p[c]; s += v * v; }
  for (int o = 16; o; o >>= 1) s += __shfl_xor(s, o, 32);
  if (lane == 0) sq[wave] = s;
}

// Gram matrix tile via f32 WMMA, LDS-shared B panel, fused distance epilogue.
// Block = 5 waves (160 threads) computing an 80(rows) x 80(cols) tile.
// Each wave owns 16 rows and all 80 cols (5 accumulators, A frag reused 5x).
// B panel Dm[j0:j0+80, k:k+32] is staged once per k-chunk in LDS and shared
// by all 5 waves: global traffic ~20 FLOP/byte instead of 4.
#define PD_JT   80
#define PD_KC   32
#define PD_STR  36          // padded LDS row stride (dwords): 16B aligned, conflict-free
#define PD_WAVES 5

__global__ __launch_bounds__(PD_WAVES * 32) void
pdist_wmma(const float* __restrict__ Dm, const float* __restrict__ sq,
           float* __restrict__ out, int n, int K) {
  __shared__ float ldsB[PD_JT * PD_STR];
  const int tid  = threadIdx.x;
  const int lane = tid & 31;
  const int wave = tid >> 5;
  const int i0 = blockIdx.x * (PD_WAVES * 16) + wave * 16;
  const int j0 = blockIdx.y * PD_JT;
  const int kh = (lane >> 4) << 1;
  const float* ap = Dm + (size_t)(i0 + (lane & 15)) * K + kh;
  v8f acc[PD_WAVES] = {};
  for (int k = 0; k < K; k += PD_KC) {
    __syncthreads();
    // cooperative stage: 80 rows x 32 k-vals = 640 float4s over 160 threads (x4)
    for (int q = tid; q < PD_JT * (PD_KC / 4); q += PD_WAVES * 32) {
      int jj = q >> 3;            // q / 8
      int kk = (q & 7) << 2;      // (q % 8) * 4
      float4 v = *(const float4*)(Dm + (size_t)(j0 + jj) * K + k + kk);
      *(float4*)&ldsB[jj * PD_STR + kk] = v;
    }
    __syncthreads();
#pragma unroll
    for (int kk = 0; kk < PD_KC; kk += 4) {
      v2f a; a[0] = ap[k + kk]; a[1] = ap[k + kk + 1];
#pragma unroll
      for (int t = 0; t < PD_WAVES; ++t) {
        const float* bp = &ldsB[(t * 16 + (lane & 15)) * PD_STR + kk + kh];
        v2f b; b[0] = bp[0]; b[1] = bp[1];
        acc[t] = __builtin_amdgcn_wmma_f32_16x16x4_f32(false, a, false, b, (short)0,
                                                       acc[t], false, false);
      }
    }
  }
  const int ir = i0 + ((lane >> 4) << 3);
  const int jb = j0 + (lane & 15);
#pragma unroll
  for (int r = 0; r < 8; ++r) {
    int i = ir + r;
    float sqi = sq[i];
    float* op = out + (size_t)i * n + jb;
#pragma unroll
    for (int t = 0; t < PD_WAVES; ++t) {
      float d2 = sqi + sq[jb + t * 16] - 2.0f * acc[t][r];
      d2 = fmaxf(d2, 0.0f);
      op[t * 16] = d2 > 0.0f ? sqrtf(d2) : 0.0f;
    }
  }
}

// ---------------------------------------------------------------- driver
extern "C" void kernel_launch(void* const* d_in, const int* in_sizes, int n_in,
                              void* d_out, int out_size, void* d_ws, size_t ws_size,
                              hipStream_t stream) {
  const float* x        = (const float*)d_in[0];
  const int*   eidx     = (const int*)d_in[1];
  const float* egat_W   = (const float*)d_in[2];
  const float* egat_as  = (const float*)d_in[3];
  const float* egat_ad  = (const float*)d_in[4];
  const float* egat_b   = (const float*)d_in[5];
  const float* egcn_W   = (const float*)d_in[6];
  const float* egcn_b   = (const float*)d_in[7];
  const float* densea_W = (const float*)d_in[8];
  const float* densea_b = (const float*)d_in[9];
  const float* latent_W = (const float*)d_in[10];
  const float* latent_b = (const float*)d_in[11];
  const float* dec1_W   = (const float*)d_in[12];
  const float* dec1_b   = (const float*)d_in[13];
  const float* dec2_W   = (const float*)d_in[14];
  const float* dec2_b   = (const float*)d_in[15];
  const float* dgcn_W   = (const float*)d_in[16];
  const float* dgcn_b   = (const float*)d_in[17];
  const float* dgat_W   = (const float*)d_in[18];
  const float* dgat_as  = (const float*)d_in[19];
  const float* dgat_ad  = (const float*)d_in[20];
  const float* dgat_b   = (const float*)d_in[21];

  const int n = in_sizes[0] / 512;          // 10000 nodes
  const int E = in_sizes[1] / 2;            // 160000 edges
  const int* src = eidx;
  const int* dst = eidx + E;

  // workspace carve-up
  float* big0 = (float*)d_ws;                       // [n,1024]
  float* big1 = big0 + (size_t)n * HC;              // [n,1024]
  float* big2 = big1 + (size_t)n * HC;              // [n,1024]
  float* alS  = big2 + (size_t)n * HC;              // [n,2]
  float* alD  = alS + (size_t)n * HEADS;
  unsigned* emaxB = (unsigned*)(alD + (size_t)n * HEADS);
  float* esum = (float*)(emaxB + (size_t)n * HEADS);
  float* dinv = esum + (size_t)n * HEADS;
  float* sq   = dinv + n;
  float* outm = (float*)d_out;

  const int EN = E + n;                             // edges incl. self-loops

  auto cdiv = [](int a, int b) { return (a + b - 1) / b; };
  auto gemm = [&](const float* A, const float* B, float* C, int M, int N, int K) {
    gemm_f32_wmma<<<dim3(M / 16, N / 64), dim3(32), 0, stream>>>(A, B, C, M, N, K);
  };
  auto fill = [&](float* p, float v, int cnt) {
    fill_f32<<<cdiv(cnt, 256), 256, 0, stream>>>(p, v, cnt);
  };
  auto bias = [&](float* X, const float* b, int cols, int relu) {
    bias_act<<<cdiv(n * cols, 256), 256, 0, stream>>>(X, b, n * cols, cols, relu);
  };
  // full GAT layer: Hm = node features already multiplied by W ([n,1024])
  auto gat = [&](const float* Hm, const float* aS, const float* aD,
                 const float* b, float* out, int relu) {
    row_dot2<<<cdiv(n * HEADS * 32, 256), 256, 0, stream>>>(Hm, aS, aD, alS, alD, n);
    fill((float*)emaxB, 0.0f, n * HEADS);   // key 0 == -inf sentinel
    gat_emax<<<cdiv(EN * HEADS, 256), 256, 0, stream>>>(src, dst, alS, alD, emaxB, E, n);
    fill(esum, 0.0f, n * HEADS);
    gat_esum<<<cdiv(EN * HEADS, 256), 256, 0, stream>>>(src, dst, alS, alD, emaxB, esum, E, n);
    fill(out, 0.0f, n * HC);
    gat_agg<<<EN, 256, 0, stream>>>(src, dst, Hm, alS, alD, emaxB, esum, out, E);
    bias(out, b, HC, relu);
  };
  auto gcn = [&](const float* Hm, const float* b, float* out, int cols, int relu) {
    fill(out, 0.0f, n * cols);
    gcn_agg<<<EN, 256, 0, stream>>>(src, dst, Hm, dinv, out, E, cols);
    bias(out, b, cols, relu);
  };

  // degree / dinv (shared by both GCN layers): init 1.0 for self loops
  fill(dinv, 1.0f, n);
  deg_accum<<<cdiv(E, 256), 256, 0, stream>>>(dst, dinv, E);
  deg_to_dinv<<<cdiv(n, 256), 256, 0, stream>>>(dinv, n);

  // ---------------- encoder ----------------
  gemm(x, egat_W, big0, n, HC, 512);                 // h = x @ W_gat   [n,1024]
  gat(big0, egat_as, egat_ad, egat_b, big1, 1);      // relu(GAT)       [n,1024]
  gemm(big1, egcn_W, big0, n, 512, HC);              // h @ W_gcn       [n,512]
  gcn(big0, egcn_b, big2, 512, 1);                   // relu(GCN)       [n,512]
  gemm(big2, densea_W, big0, n, 128, 512);           // densea          [n,128]
  bias(big0, densea_b, 128, 1);
  gemm(big0, latent_W, big1, n, 64, 128);            // latent z        [n,64]
  bias(big1, latent_b, 64, 0);

  // ---------------- decoder ----------------
  gemm(big1, dec1_W, big2, n, 128, 64);              // dec1            [n,128]
  bias(big2, dec1_b, 128, 1);
  gemm(big2, dec2_W, big0, n, 512, 128);             // dec2            [n,512]
  bias(big0, dec2_b, 512, 1);
  gemm(big0, dgcn_W, big1, n, 512, 512);             // h @ W_gcn       [n,512]
  gcn(big1, dgcn_b, big2, 512, 1);                   // relu(GCN)       [n,512]
  gemm(big2, dgat_W, big0, n, HC, 512);              // h = d @ W_gat   [n,1024]
  gat(big0, dgat_as, dgat_ad, dgat_b, big1, 0);      // GAT (no relu)   [n,1024]

  // ---------------- pdist ----------------
  rowsq<<<cdiv(n * 32, 256), 256, 0, stream>>>(big1, sq, n, HC);
  pdist_wmma<<<dim3(n / PD_JT, n / PD_JT), dim3(PD_WAVES * 32), 0, stream>>>(big1, sq, outm, n, HC);
}